// GCNNet_69097433858684
// MI455X (gfx1250) — compile-verified
//
#include <hip/hip_runtime.h>
#include <hip/hip_bf16.h>

typedef __attribute__((ext_vector_type(2))) float v2f;
typedef __attribute__((ext_vector_type(8))) float v8f;

#define F_IN 512
#define HID  16
#define COUT 40

// Guaranteed hardware float atomic: non-returning GLOBAL_ATOMIC_ADD_F32.
// (atomicAdd(float*) can lower to a CAS loop; this cannot.)
__device__ __forceinline__ void atomic_add_f32_hw(float* p, float v) {
    asm volatile("global_atomic_add_f32 %0, %1, off"
                 :
                 : "v"(p), "v"(v)
                 : "memory");
}

// ---------------------------------------------------------------------------
// Zero-fill (agg buffers + degree counters must be zero every call)
// ---------------------------------------------------------------------------
__global__ void zero_kernel(float* __restrict__ p, long n) {
    long i = (long)blockIdx.x * blockDim.x + threadIdx.x;
    long stride = (long)gridDim.x * blockDim.x;
    for (; i < n; i += stride) p[i] = 0.0f;
}

// ---------------------------------------------------------------------------
// Degree accumulation: out_deg[src[e]]++, in_deg[dst[e]]++
// ---------------------------------------------------------------------------
__global__ void degree_kernel(const int* __restrict__ src, const int* __restrict__ dst,
                              int* __restrict__ out_deg, int* __restrict__ in_deg, int E) {
    int e = blockIdx.x * blockDim.x + threadIdx.x;
    if (e >= E) return;
    atomicAdd(&out_deg[src[e]], 1);
    atomicAdd(&in_deg[dst[e]], 1);
}

// ---------------------------------------------------------------------------
// norm = rsqrt(clip(deg, 1))
// ---------------------------------------------------------------------------
__global__ void norm_kernel(const int* __restrict__ out_deg, const int* __restrict__ in_deg,
                            float* __restrict__ norm_src, float* __restrict__ norm_dst, int N) {
    int i = blockIdx.x * blockDim.x + threadIdx.x;
    if (i >= N) return;
    norm_src[i] = rsqrtf(fmaxf((float)out_deg[i], 1.0f));
    norm_dst[i] = rsqrtf(fmaxf((float)in_deg[i], 1.0f));
}

// ---------------------------------------------------------------------------
// GEMM1: h1[N,16] = (x * norm_src[:,None]) @ W1   (f32 WMMA 16x16x4)
// One wave per 16-row tile; 8 tiles per 256-thread block; W1 staged in LDS.
// ---------------------------------------------------------------------------
__global__ __launch_bounds__(256) void gemm1_kernel(
    const float* __restrict__ x, const float* __restrict__ W1,
    const float* __restrict__ norm_src, float* __restrict__ h1, int nTiles)
{
    __shared__ float Wlds[F_IN * HID];  // 32 KB
    for (int i = threadIdx.x; i < F_IN * HID; i += blockDim.x) Wlds[i] = W1[i];
    __syncthreads();

    const int wave = threadIdx.x >> 5;
    const int lane = threadIdx.x & 31;
    const int tile = blockIdx.x * 8 + wave;
    if (tile >= nTiles) return;      // wave-uniform: EXEC stays all-1s for WMMA

    const int hi  = lane >> 4;       // half-wave select (K pair)
    const int col = lane & 15;       // A: M row index; B/C/D: N col index
    const long row = (long)tile * 16 + col;   // N % 16 == 0, no masking needed
    const float rn = norm_src[row];
    const float* xrow = x + row * F_IN;

    v8f c = {0.f, 0.f, 0.f, 0.f, 0.f, 0.f, 0.f, 0.f};
    #pragma unroll 8
    for (int k = 0; k < F_IN; k += 4) {
        const int kb = k + 2 * hi;   // even -> 8B aligned
        v2f a = *(const v2f*)(xrow + kb);
        a.x *= rn; a.y *= rn;
        v2f b;
        b.x = Wlds[kb * HID + col];
        b.y = Wlds[(kb + 1) * HID + col];
        c = __builtin_amdgcn_wmma_f32_16x16x4_f32(
                /*neg_a=*/false, a, /*neg_b=*/false, b,
                /*c_mod=*/(short)0, c, /*reuse_a=*/false, /*reuse_b=*/false);
    }

    float* out = h1 + (long)tile * 16 * HID;
    #pragma unroll
    for (int r = 0; r < 8; ++r)
        out[(r + 8 * hi) * HID + col] = c[r];   // element (M = r+8*hi, N = col)
}

// ---------------------------------------------------------------------------
// Edge scatter-add, 16 features/edge: agg[dst[e],f] += feat[src[e],f]
// 16 consecutive lanes share one edge -> 64B coalesced gather, and the 16
// atomics of one edge hit consecutive addresses (single cacheline at L2).
// ---------------------------------------------------------------------------
__global__ void edge_scatter16_kernel(const int* __restrict__ src, const int* __restrict__ dst,
                                      const float* __restrict__ feat, float* __restrict__ agg, int E) {
    int tid = blockIdx.x * blockDim.x + threadIdx.x;
    int e = tid >> 4;
    if (e >= E) return;
    int f = tid & 15;
    float v = feat[(long)src[e] * HID + f];
    atomic_add_f32_hw(&agg[(long)dst[e] * HID + f], v);
}

// ---------------------------------------------------------------------------
// s = relu(agg1 * norm_dst + b1) * norm_src   (fused layer-1 epilogue +
// layer-2 source normalization; s is the 16-wide message for layer 2)
// ---------------------------------------------------------------------------
__global__ void relu_scale_kernel(const float* __restrict__ agg1,
                                  const float* __restrict__ norm_src,
                                  const float* __restrict__ norm_dst,
                                  const float* __restrict__ b1,
                                  float* __restrict__ s, long n16) {
    long i = (long)blockIdx.x * blockDim.x + threadIdx.x;
    if (i >= n16) return;
    long row = i >> 4;
    int  f   = (int)(i & 15);
    float v = fmaf(agg1[i], norm_dst[row], b1[f]);
    v = fmaxf(v, 0.0f);
    s[i] = v * norm_src[row];
}

// ---------------------------------------------------------------------------
// GEMM2: out[N,40] = (agg2 * norm_dst[:,None]) @ W2 + b2
// W2 zero-padded to 16x48 in LDS; 3 WMMA col-tiles, last store masked.
// ---------------------------------------------------------------------------
__global__ __launch_bounds__(256) void gemm2_kernel(
    const float* __restrict__ agg2, const float* __restrict__ W2,
    const float* __restrict__ b2, const float* __restrict__ norm_dst,
    float* __restrict__ out, int nTiles)
{
    __shared__ float Wlds[HID * 48];
    __shared__ float blds[48];
    for (int i = threadIdx.x; i < HID * 48; i += blockDim.x) {
        int r = i / 48, cc = i % 48;
        Wlds[i] = (cc < COUT) ? W2[r * COUT + cc] : 0.0f;
    }
    if (threadIdx.x < 48) blds[threadIdx.x] = (threadIdx.x < COUT) ? b2[threadIdx.x] : 0.0f;
    __syncthreads();

    const int wave = threadIdx.x >> 5;
    const int lane = threadIdx.x & 31;
    const int tile = blockIdx.x * 8 + wave;
    if (tile >= nTiles) return;

    const int hi  = lane >> 4;
    const int col = lane & 15;
    const long arow = (long)tile * 16 + col;
    const float rn = norm_dst[arow];
    const float* A = agg2 + arow * HID;

    v2f afrag[4];
    #pragma unroll
    for (int kk = 0; kk < 4; ++kk) {
        const int kb = kk * 4 + 2 * hi;
        v2f a = *(const v2f*)(A + kb);
        a.x *= rn; a.y *= rn;
        afrag[kk] = a;
    }

    float* orow = out + (long)tile * 16 * COUT;
    #pragma unroll
    for (int t = 0; t < 3; ++t) {
        v8f c = {0.f, 0.f, 0.f, 0.f, 0.f, 0.f, 0.f, 0.f};
        #pragma unroll
        for (int kk = 0; kk < 4; ++kk) {
            const int kb = kk * 4 + 2 * hi;
            v2f b;
            b.x = Wlds[kb * 48 + t * 16 + col];
            b.y = Wlds[(kb + 1) * 48 + t * 16 + col];
            c = __builtin_amdgcn_wmma_f32_16x16x4_f32(
                    false, afrag[kk], false, b, (short)0, c, false, false);
        }
        const int ccol = t * 16 + col;
        if (ccol < COUT) {   // store mask only; WMMA above ran with full EXEC
            #pragma unroll
            for (int r = 0; r < 8; ++r)
                orow[(r + 8 * hi) * COUT + ccol] = c[r] + blds[ccol];
        }
    }
}

// ---------------------------------------------------------------------------
// Host orchestration
// ---------------------------------------------------------------------------
extern "C" void kernel_launch(void* const* d_in, const int* in_sizes, int n_in,
                              void* d_out, int out_size, void* d_ws, size_t ws_size,
                              hipStream_t stream) {
    const float* x   = (const float*)d_in[0];
    const int* esrc  = (const int*)d_in[1];
    const int* edst  = (const int*)d_in[2];
    const float* W1  = (const float*)d_in[3];
    const float* b1  = (const float*)d_in[4];
    const float* W2  = (const float*)d_in[5];
    const float* b2  = (const float*)d_in[6];
    float* out = (float*)d_out;

    const long N = in_sizes[0] / F_IN;     // 100000
    const int  E = in_sizes[1];            // 3200000

    // Workspace layout (floats). Zeroed region first: [out_deg|in_deg|agg1|agg2]
    float* ws = (float*)d_ws;
    int*   out_deg  = (int*)ws;            // N
    int*   in_deg   = (int*)(ws + N);      // N
    float* agg1     = ws + 2 * N;          // 16N
    float* agg2     = ws + 18 * N;         // 16N
    float* norm_src = ws + 34 * N;         // N
    float* norm_dst = ws + 35 * N;         // N
    float* h1       = ws + 36 * N;         // 16N
    float* s        = ws + 52 * N;         // 16N   (total 68N floats ~ 27.2 MB)

    const int BLK = 256;
    const long nZero = 34 * N;
    int zgrid = (int)((nZero + BLK - 1) / BLK);
    if (zgrid > 65535) zgrid = 65535;      // grid-stride
    zero_kernel<<<zgrid, BLK, 0, stream>>>(ws, nZero);

    degree_kernel<<<(E + BLK - 1) / BLK, BLK, 0, stream>>>(esrc, edst, out_deg, in_deg, E);
    norm_kernel<<<(int)((N + BLK - 1) / BLK), BLK, 0, stream>>>(out_deg, in_deg, norm_src, norm_dst, (int)N);

    const int nTiles = (int)(N / 16);                     // N % 16 == 0
    const int gBlocks = (nTiles + 7) / 8;
    gemm1_kernel<<<gBlocks, BLK, 0, stream>>>(x, W1, norm_src, h1, nTiles);

    const int eThreads16 = E * 16;
    edge_scatter16_kernel<<<(eThreads16 + BLK - 1) / BLK, BLK, 0, stream>>>(esrc, edst, h1, agg1, E);

    const long n16 = N * 16;
    relu_scale_kernel<<<(int)((n16 + BLK - 1) / BLK), BLK, 0, stream>>>(agg1, norm_src, norm_dst, b1, s, n16);

    edge_scatter16_kernel<<<(eThreads16 + BLK - 1) / BLK, BLK, 0, stream>>>(esrc, edst, s, agg2, E);

    gemm2_kernel<<<gBlocks, BLK, 0, stream>>>(agg2, W2, b2, norm_dst, out, nTiles);
}